// SpanMaxPooler_24515673325655
// MI455X (gfx1250) — compile-verified
//
#include <hip/hip_runtime.h>
#include <cfloat>

// SpanMaxPooler for MI455X (gfx1250).
// B=32, S=4096, H=1024, K=2. Output (B, K*H) fp32.
// Streaming max-reduction: bandwidth-bound (~16.8 MB @ 23.3 TB/s ~ 0.7us).
// Uses the CDNA5 async global->LDS DMA path (ASYNCcnt) as a 2-deep
// double-buffered pipeline feeding v_max_num_f32 from LDS.

#define AS3 __attribute__((address_space(3)))

namespace {
constexpr int Bc = 32;
constexpr int Sc = 4096;
constexpr int Hc = 1024;
constexpr int Kc = 2;

constexpr int CHUNKS      = 4;               // H split into 4 chunks of 256
constexpr int CHUNK_F     = Hc / CHUNKS;     // 256 floats per chunk
constexpr int THREADS     = CHUNK_F / 4;     // 64 threads, one float4 each
constexpr int STAGE_ROWS  = 8;               // rows DMA'd per pipeline stage
constexpr int NBUF        = 2;               // double buffer
}

__global__ __launch_bounds__(THREADS)
void span_max_pool_kernel(const float* __restrict__ hidden,
                          const int*   __restrict__ start_idx,
                          const int*   __restrict__ end_idx,
                          const float* __restrict__ missing,
                          float*       __restrict__ out)
{
    // 2 stages * 8 rows * 256 floats = 16 KB LDS
    __shared__ __align__(16) float lds_buf[NBUF * STAGE_ROWS * CHUNK_F];

    const int t     = threadIdx.x;        // 0..63
    const int bk    = blockIdx.x;         // 0..63
    const int chunk = blockIdx.y;         // 0..3
    const int b     = bk >> 1;
    const int k     = bk & 1;

    const int cc = t * 4;                 // column within chunk (bytes/4)
    const int c0 = chunk * CHUNK_F + cc;  // absolute feature column

    const int st = start_idx[b * Kc + k];
    const int en = end_idx[b * Kc + k];

    float* outp = out + ((size_t)b * (Kc * Hc) + (size_t)k * Hc + c0);

    // Missing span: uniform per block -> EXEC stays full either way.
    if (st < 0 || en < 0) {
        const float4 me = *(const float4*)(missing + (size_t)k * Hc + c0);
        *(float4*)outp = me;
        return;
    }

    const float* hb = hidden + (size_t)b * Sc * Hc + c0;  // this thread's column
    const unsigned lds_base = (unsigned)(unsigned long long)(AS3 float*)lds_buf;

    const int len  = en - st;                              // >= 1
    const int nst  = (len + STAGE_ROWS - 1) / STAGE_ROWS;  // pipeline stages

    // Issue one stage: 8 async b128 loads per wave (rows clamped to en-1 so
    // duplicates are idempotent under max; keeps issue uniform/branch-free).
    auto issue_stage = [&](int s, int buf) {
        // WAR guard: make sure our pending ds reads of this buffer are done
        // before the DMA engine overwrites it.
        asm volatile("s_wait_dscnt 0x0" ::: "memory");
        #pragma unroll
        for (int r = 0; r < STAGE_ROWS; ++r) {
            int row = st + s * STAGE_ROWS + r;
            row = (row < en) ? row : (en - 1);
            const float* gp = hb + (size_t)row * Hc;
            unsigned long long ga = (unsigned long long)gp;
            unsigned loff = lds_base +
                (unsigned)(((buf * STAGE_ROWS + r) * CHUNK_F + cc) * 4);
            asm volatile("global_load_async_to_lds_b128 %0, %1, off"
                         :: "v"(loff), "v"(ga)
                         : "memory");
        }
    };

    issue_stage(0, 0);

    float4 acc = make_float4(-FLT_MAX, -FLT_MAX, -FLT_MAX, -FLT_MAX);

    for (int s = 0; s < nst; ++s) {
        const int buf = s & 1;
        if (s + 1 < nst) {
            issue_stage(s + 1, buf ^ 1);
            // Async loads complete in order: <=8 outstanding means the 8
            // loads of stage s have landed in LDS.
            asm volatile("s_wait_asynccnt 8" ::: "memory");
        } else {
            asm volatile("s_wait_asynccnt 0x0" ::: "memory");
        }
        #pragma unroll
        for (int r = 0; r < STAGE_ROWS; ++r) {
            const float4 v =
                *(const float4*)&lds_buf[(buf * STAGE_ROWS + r) * CHUNK_F + cc];
            acc.x = fmaxf(acc.x, v.x);
            acc.y = fmaxf(acc.y, v.y);
            acc.z = fmaxf(acc.z, v.z);
            acc.w = fmaxf(acc.w, v.w);
        }
    }

    *(float4*)outp = acc;
}

extern "C" void kernel_launch(void* const* d_in, const int* in_sizes, int n_in,
                              void* d_out, int out_size, void* d_ws, size_t ws_size,
                              hipStream_t stream) {
    (void)in_sizes; (void)n_in; (void)out_size; (void)d_ws; (void)ws_size;
    const float* hidden  = (const float*)d_in[0];
    const int*   start_i = (const int*)  d_in[1];
    const int*   end_i   = (const int*)  d_in[2];
    const float* missing = (const float*)d_in[3];
    float*       out     = (float*)d_out;

    dim3 grid(Bc * Kc, CHUNKS);   // 64 x 4 = 256 workgroups
    dim3 block(THREADS);          // 64 threads = 2 waves
    span_max_pool_kernel<<<grid, block, 0, stream>>>(hidden, start_i, end_i,
                                                     missing, out);
}